// TaskTaskGAT2Layer_73873437491428
// MI455X (gfx1250) — compile-verified
//
#include <hip/hip_runtime.h>
#include <hip/hip_bf16.h>
#include <math.h>

typedef __attribute__((ext_vector_type(2))) float v2f;
typedef __attribute__((ext_vector_type(8))) float v8f;

#define ATT_NEG 0.2f
#define LN_EPS 1e-5f
#define SM_EPS 1e-16f

static inline int cdiv(int a, int b) { return (a + b - 1) / b; }

// ---------------------------------------------------------------------------
// Float atomic max via int max / uint min bit trick (valid with -inf init).
// ---------------------------------------------------------------------------
__device__ __forceinline__ void atomicMaxF(float* addr, float v) {
  if (v >= 0.0f) atomicMax((int*)addr, __float_as_int(v));
  else           atomicMin((unsigned int*)addr, __float_as_uint(v));
}

// ---------------------------------------------------------------------------
// Y[m, n] = sum_k X[m,k] * W[n,k] (+ bias[n])  via V_WMMA_F32_16X16X4_F32.
// X: [nrows, K] row-major. W: [O, K] row-major. Y: ldY stride (col base via ptr).
// One wave (32 threads) computes one 16x16 tile. K must be a multiple of 4.
// A frag (16x4 f32): lane m = l&15; VGPR v holds K = k0 + v + (l>=16 ? 2 : 0).
// B frag (4x16 f32): lane n = l&15; VGPR v holds K = k0 + v + (l>=16 ? 2 : 0).
// C/D (16x16 f32): VGPR v: lanes 0-15 -> M=v, lanes 16-31 -> M=v+8; N = l&15.
// ---------------------------------------------------------------------------
__global__ void k_linear_wmma(const float* __restrict__ X, const float* __restrict__ W,
                              const float* __restrict__ bias, float* __restrict__ Y,
                              int nrows, int K, int ldY) {
  const int l    = threadIdx.x;
  const int half = l >> 4;
  const int lm   = l & 15;
  const int m0   = blockIdx.x << 4;
  const int n0   = blockIdx.y << 4;
  int row = m0 + lm;
  if (row >= nrows) row = nrows - 1;          // clamp (WMMA needs full EXEC)
  const float* xr = X + (size_t)row * K;
  const float* wr = W + (size_t)(n0 + lm) * K;

  v8f acc = {0.f, 0.f, 0.f, 0.f, 0.f, 0.f, 0.f, 0.f};
  for (int k0 = 0; k0 < K; k0 += 4) {
    const int ka = k0 + (half << 1);
    v2f a, b;
    a.x = xr[ka];  a.y = xr[ka + 1];
    b.x = wr[ka];  b.y = wr[ka + 1];
    acc = __builtin_amdgcn_wmma_f32_16x16x4_f32(false, a, false, b,
                                                (short)0, acc, false, false);
  }
  const float bv = bias ? bias[n0 + lm] : 0.f;
#pragma unroll
  for (int v = 0; v < 8; ++v) {
    const int orow = m0 + v + (half << 3);
    if (orow < nrows) Y[(size_t)orow * ldY + (n0 + lm)] = acc[v] + bv;
  }
}

// ---------------------------------------------------------------------------
__global__ void k_fill(float* p, float v, int n) {
  int i = blockIdx.x * blockDim.x + threadIdx.x;
  if (i < n) p[i] = v;
}

// coef[h] = sum_c W_edge[h*ch+c] * att_edge[h*ch+c]   (ED == 1)
__global__ void k_edge_coef(const float* __restrict__ W_edge,
                            const float* __restrict__ att_edge,
                            float* coef, int heads, int ch) {
  int h = threadIdx.x;
  if (h >= heads) return;
  float s = 0.f;
  for (int c = 0; c < ch; ++c) s += W_edge[h * ch + c] * att_edge[h * ch + c];
  coef[h] = s;
}

// a_src[i,h] = dot(xs[i,h,:], att_src[h,:]);  a_dst likewise
__global__ void k_node_alpha(const float* __restrict__ xs, const float* __restrict__ xd,
                             const float* __restrict__ att_src, const float* __restrict__ att_dst,
                             float* a_src, float* a_dst, int n, int heads, int ch) {
  int i = blockIdx.x * blockDim.x + threadIdx.x;
  if (i >= n * heads) return;
  int node = i / heads, h = i - node * heads;
  const float* ps = xs + (size_t)node * heads * ch + h * ch;
  const float* pd = xd + (size_t)node * heads * ch + h * ch;
  float s1 = 0.f, s2 = 0.f;
  for (int c = 0; c < ch; ++c) {
    s1 += ps[c] * att_src[h * ch + c];
    s2 += pd[c] * att_dst[h * ch + c];
  }
  a_src[i] = s1;
  a_dst[i] = s2;
}

// alpha = leaky_relu(a_src[s] + a_dst[d] + eattr*coef, 0.2); segment max into amax
__global__ void k_edge_alpha(const int* __restrict__ src, const int* __restrict__ dst,
                             const float* __restrict__ eattr,
                             const float* __restrict__ a_src, const float* __restrict__ a_dst,
                             const float* __restrict__ coef,
                             float* alpha, float* amax, int nE, int heads) {
  int i = blockIdx.x * blockDim.x + threadIdx.x;
  if (i >= nE * heads) return;
  int e = i / heads, h = i - e * heads;
  int s = src[e], d = dst[e];
  float a = a_src[s * heads + h] + a_dst[d * heads + h] + eattr[e] * coef[h];
  a = (a > 0.f) ? a : ATT_NEG * a;
  alpha[i] = a;
  atomicMaxF(&amax[d * heads + h], a);
}

__global__ void k_fix_amax(float* amax, int n) {
  int i = blockIdx.x * blockDim.x + threadIdx.x;
  if (i >= n) return;
  float a = amax[i];
  amax[i] = isfinite(a) ? a : 0.f;
}

// ex = exp(alpha - amax[d]); denom[d] += ex (in-place on alpha buffer)
__global__ void k_edge_exp(const int* __restrict__ dst, float* alpha_ex,
                           const float* __restrict__ amax, float* denom,
                           int nE, int heads) {
  int i = blockIdx.x * blockDim.x + threadIdx.x;
  if (i >= nE * heads) return;
  int e = i / heads, h = i - e * heads;
  int d = dst[e];
  float ex = expf(alpha_ex[i] - amax[d * heads + h]);
  alpha_ex[i] = ex;
  atomicAdd(&denom[d * heads + h], ex);
}

// out[d, col] += xs[s,h,c] * ex/(denom+eps) * scale ; col = concat ? h*ch+c : c
__global__ void k_edge_aggregate(const int* __restrict__ src, const int* __restrict__ dst,
                                 const float* __restrict__ xs, const float* __restrict__ ex,
                                 const float* __restrict__ denom, float* out,
                                 int nE, int heads, int ch, int concat, int ldout, float scale) {
  int i = blockIdx.x * blockDim.x + threadIdx.x;
  int total = nE * heads * ch;
  if (i >= total) return;
  int e = i / (heads * ch);
  int r = i - e * heads * ch;
  int h = r / ch;
  int c = r - h * ch;
  int s = src[e], d = dst[e];
  float w   = ex[e * heads + h] / (denom[d * heads + h] + SM_EPS);
  float val = xs[(size_t)s * heads * ch + r] * w * scale;
  int col = concat ? r : c;
  atomicAdd(&out[(size_t)d * ldout + col], val);
}

// LayerNorm over 32 channels then leaky_relu(0.01)
__global__ void k_ln_lrelu(const float* __restrict__ x, const float* __restrict__ g,
                           const float* __restrict__ b, float* __restrict__ y, int n) {
  int i = blockIdx.x * blockDim.x + threadIdx.x;
  if (i >= n) return;
  const float* row = x + (size_t)i * 32;
  float mu = 0.f;
#pragma unroll
  for (int c = 0; c < 32; ++c) mu += row[c];
  mu *= (1.f / 32.f);
  float var = 0.f;
#pragma unroll
  for (int c = 0; c < 32; ++c) { float dv = row[c] - mu; var += dv * dv; }
  var *= (1.f / 32.f);
  float inv = rsqrtf(var + LN_EPS);
#pragma unroll
  for (int c = 0; c < 32; ++c) {
    float v = (row[c] - mu) * inv * g[c] + b[c];
    y[(size_t)i * 32 + c] = (v > 0.f) ? v : 0.01f * v;
  }
}

// self-loop helpers for dpt2 (PyG add_self_loops with fill_value='mean')
__global__ void k_count_sum(const int* __restrict__ fdst, const float* __restrict__ eattr,
                            float* cnt, float* asum, int nE) {
  int i = blockIdx.x * blockDim.x + threadIdx.x;
  if (i >= nE) return;
  atomicAdd(&cnt[fdst[i]], 1.f);
  atomicAdd(&asum[fdst[i]], eattr[i]);
}

__global__ void k_attr_self(const float* cnt, const float* asum, float* attr_self, int n) {
  int i = blockIdx.x * blockDim.x + threadIdx.x;
  if (i >= n) return;
  attr_self[i] = (cnt[i] > 0.f) ? asum[i] / cnt[i] : 0.f;
}

__global__ void k_build_edges2(const int* __restrict__ fsrc, const int* __restrict__ fdst,
                               const float* __restrict__ eattr, const float* __restrict__ attr_self,
                               int* s2, int* d2, float* ea2, int nE, int n) {
  int i = blockIdx.x * blockDim.x + threadIdx.x;
  if (i >= nE + n) return;
  if (i < nE) { s2[i] = fsrc[i]; d2[i] = fdst[i]; ea2[i] = eattr[i]; }
  else { int v = i - nE; s2[i] = v; d2[i] = v; ea2[i] = attr_self[v]; }
}

// ---------------------------------------------------------------------------
struct Gat {
  const float *W_src, *W_dst, *W_edge, *att_src, *att_dst, *att_edge, *W_res, *bias;
};

static void run_conv(const float* x, int f_in, const int* src, const int* dst,
                     const float* eattr, int nE, const Gat& P, int heads, int ch, bool concat,
                     float* out, int ldout,
                     float* xs, float* xd, float* a_src, float* a_dst,
                     float* amax, float* denom, float* ex, float* coef,
                     int n, hipStream_t stream) {
  const int HC   = heads * ch;
  const int outw = concat ? HC : ch;      // 32 in all convs here
  dim3 blk32(32, 1, 1);
  k_linear_wmma<<<dim3(cdiv(n, 16), HC / 16), blk32, 0, stream>>>(x, P.W_src, nullptr, xs, n, f_in, HC);
  k_linear_wmma<<<dim3(cdiv(n, 16), HC / 16), blk32, 0, stream>>>(x, P.W_dst, nullptr, xd, n, f_in, HC);
  // residual + bias pre-fills the output, messages get atomically added on top
  k_linear_wmma<<<dim3(cdiv(n, 16), outw / 16), blk32, 0, stream>>>(x, P.W_res, P.bias, out, n, f_in, ldout);
  k_edge_coef<<<1, 32, 0, stream>>>(P.W_edge, P.att_edge, coef, heads, ch);

  const int nh = n * heads;
  k_node_alpha<<<cdiv(nh, 256), 256, 0, stream>>>(xs, xd, P.att_src, P.att_dst, a_src, a_dst, n, heads, ch);
  k_fill<<<cdiv(nh, 256), 256, 0, stream>>>(amax, -INFINITY, nh);
  k_fill<<<cdiv(nh, 256), 256, 0, stream>>>(denom, 0.f, nh);

  const int eh = nE * heads;
  k_edge_alpha<<<cdiv(eh, 256), 256, 0, stream>>>(src, dst, eattr, a_src, a_dst, coef, ex, amax, nE, heads);
  k_fix_amax<<<cdiv(nh, 256), 256, 0, stream>>>(amax, nh);
  k_edge_exp<<<cdiv(eh, 256), 256, 0, stream>>>(dst, ex, amax, denom, nE, heads);

  const int ehc = nE * heads * ch;
  const float scale = concat ? 1.f : 1.f / (float)heads;
  k_edge_aggregate<<<cdiv(ehc, 256), 256, 0, stream>>>(src, dst, xs, ex, denom, out,
                                                       nE, heads, ch, (int)concat, ldout, scale);
}

extern "C" void kernel_launch(void* const* d_in, const int* in_sizes, int n_in,
                              void* d_out, int out_size, void* d_ws, size_t ws_size,
                              hipStream_t stream) {
  const float* x     = (const float*)d_in[0];
  const int*   ei    = (const int*)d_in[1];
  const float* eattr = (const float*)d_in[2];
  const int N_ = in_sizes[0] / 12;   // 100000
  const int E_ = in_sizes[1] / 2;    // 1600000
  const int* src = ei;               // edge_index[0]
  const int* dst = ei + E_;          // edge_index[1]

  auto gp = [&](int base) {
    Gat g;
    g.W_src    = (const float*)d_in[base + 0];
    g.W_dst    = (const float*)d_in[base + 1];
    g.W_edge   = (const float*)d_in[base + 2];
    g.att_src  = (const float*)d_in[base + 3];
    g.att_dst  = (const float*)d_in[base + 4];
    g.att_edge = (const float*)d_in[base + 5];
    g.W_res    = (const float*)d_in[base + 6];
    g.bias     = (const float*)d_in[base + 7];
    return g;
  };
  Gat dep1 = gp(3), dpt1 = gp(11), dep2 = gp(19), dpt2 = gp(27);
  const float* ln_dep_g = (const float*)d_in[35];
  const float* ln_dep_b = (const float*)d_in[36];
  const float* ln_dpt_g = (const float*)d_in[37];
  const float* ln_dpt_b = (const float*)d_in[38];
  float* out = (float*)d_out;

  // carve workspace (256B aligned chunks)
  char* wsb = (char*)d_ws;
  size_t off = 0;
  auto carve = [&](size_t elems) -> void* {
    void* p = wsb + off;
    off += ((elems * 4 + 255) / 256) * 256;
    return p;
  };
  float* xs      = (float*)carve((size_t)N_ * 64);
  float* xd      = (float*)carve((size_t)N_ * 64);
  float* a_src   = (float*)carve((size_t)N_ * 2);
  float* a_dst   = (float*)carve((size_t)N_ * 2);
  float* amax    = (float*)carve((size_t)N_ * 2);
  float* denom   = (float*)carve((size_t)N_ * 2);
  float* ex      = (float*)carve((size_t)(E_ + N_) * 2);
  float* coef    = (float*)carve(32);
  float* cnt     = (float*)carve(N_);
  float* asum    = (float*)carve(N_);
  float* attrs   = (float*)carve(N_);
  int*   s2      = (int*)carve(E_ + N_);
  int*   d2      = (int*)carve(E_ + N_);
  float* ea2     = (float*)carve(E_ + N_);
  float* h1      = (float*)carve((size_t)N_ * 32);
  float* c1out   = (float*)carve((size_t)N_ * 32);
  (void)ws_size; (void)n_in; (void)out_size;

  // ---- dependency path: edges as-is ----
  run_conv(x, 12, src, dst, eattr, E_, dep1, 2, 32, false, c1out, 32,
           xs, xd, a_src, a_dst, amax, denom, ex, coef, N_, stream);
  k_ln_lrelu<<<cdiv(N_, 256), 256, 0, stream>>>(c1out, ln_dep_g, ln_dep_b, h1, N_);
  run_conv(h1, 32, src, dst, eattr, E_, dep2, 1, 32, true, out, 64,
           xs, xd, a_src, a_dst, amax, denom, ex, coef, N_, stream);

  // ---- dependant path: flipped edges (fsrc=dst, fdst=src) ----
  run_conv(x, 12, dst, src, eattr, E_, dpt1, 2, 32, false, c1out, 32,
           xs, xd, a_src, a_dst, amax, denom, ex, coef, N_, stream);
  k_ln_lrelu<<<cdiv(N_, 256), 256, 0, stream>>>(c1out, ln_dpt_g, ln_dpt_b, h1, N_);

  // dpt2: add self loops with fill_value='mean' over fdst (= src array)
  k_fill<<<cdiv(N_, 256), 256, 0, stream>>>(cnt, 0.f, N_);
  k_fill<<<cdiv(N_, 256), 256, 0, stream>>>(asum, 0.f, N_);
  k_count_sum<<<cdiv(E_, 256), 256, 0, stream>>>(src, eattr, cnt, asum, E_);
  k_attr_self<<<cdiv(N_, 256), 256, 0, stream>>>(cnt, asum, attrs, N_);
  k_build_edges2<<<cdiv(E_ + N_, 256), 256, 0, stream>>>(dst, src, eattr, attrs, s2, d2, ea2, E_, N_);
  run_conv(h1, 32, s2, d2, ea2, E_ + N_, dpt2, 1, 32, true, out + 32, 64,
           xs, xd, a_src, a_dst, amax, denom, ex, coef, N_, stream);
}